// ProtoSiamese_46196668236116
// MI455X (gfx1250) — compile-verified
//
#include <hip/hip_runtime.h>
#include <math.h>

#define B_ 64
#define C_ 512
#define H_ 14
#define W_ 14
#define HW_ 196
#define P_ 2000
#define EPS_ 1e-8f

typedef __attribute__((ext_vector_type(2))) float v2f;
typedef __attribute__((ext_vector_type(8))) float v8f;

// ---------------------------------------------------------------------------
// Phase 1: per-(b,p) argmax over 196 sim values, both streams.
// One wave32 per row; float4 loads (196 = 49 * float4); butterfly reduce with
// first-occurrence tie-break (lower flat index wins on equal value).
// ---------------------------------------------------------------------------
__global__ void __launch_bounds__(256)
pm_argmax(const float* __restrict__ simA, const float* __restrict__ simB,
          int* __restrict__ idxA, int* __restrict__ idxB,
          float* __restrict__ outIA, float* __restrict__ outJA,
          float* __restrict__ outIB, float* __restrict__ outJB) {
  const int wave = threadIdx.x >> 5;
  const int lane = threadIdx.x & 31;
  const long long NROW = (long long)B_ * P_;
  long long r = (long long)blockIdx.x * 8 + wave;
  if (r >= 2 * NROW) return;
  const int strm = (r >= NROW);
  const long long rr = strm ? (r - NROW) : r;
  const float4* row4 = (const float4*)((strm ? simB : simA) + rr * HW_);

  float best = -INFINITY;
  int bidx = 0x7fffffff;
  // chunks 0..48; lane takes chunk lane and lane+32 (lane < 17)
  for (int c = lane; c < 49; c += 32) {
    float4 v = row4[c];
    int i0 = 4 * c;
    if (v.x > best) { best = v.x; bidx = i0; }
    if (v.y > best) { best = v.y; bidx = i0 + 1; }
    if (v.z > best) { best = v.z; bidx = i0 + 2; }
    if (v.w > best) { best = v.w; bidx = i0 + 3; }
  }
  for (int off = 16; off >= 1; off >>= 1) {
    float ov = __shfl_xor(best, off, 32);
    int oi = __shfl_xor(bidx, off, 32);
    if (ov > best || (ov == best && oi < bidx)) { best = ov; bidx = oi; }
  }
  if (lane == 0) {
    float fi = (float)(bidx / W_);
    float fj = (float)(bidx % W_);
    if (strm) { idxB[rr] = bidx; outIB[rr] = fi; outJB[rr] = fj; }
    else      { idxA[rr] = bidx; outIA[rr] = fi; outJA[rr] = fj; }
  }
}

// ---------------------------------------------------------------------------
// Phase 2: LDS-tiled transpose feat[b][c][hw] -> featT[b][hw][c] so that each
// prototype gather becomes a contiguous 2KB read (vs 512 strided DWORDs).
// grid = (ceil(196/32)=7, 512/32=16, 2*64), block = (32,8)
// ---------------------------------------------------------------------------
__global__ void __launch_bounds__(256)
pm_transpose(const float* __restrict__ featA, const float* __restrict__ featB,
             float* __restrict__ tA, float* __restrict__ tB) {
  __shared__ float tile[32][33];
  const int bz = blockIdx.z;      // stream*64 + b
  const int strm = bz >> 6;
  const int b = bz & 63;
  const float* src = (strm ? featB : featA) + (long long)b * C_ * HW_;
  float* dst = (strm ? tB : tA) + (long long)b * HW_ * C_;
  const int hw0 = blockIdx.x * 32;
  const int c0 = blockIdx.y * 32;
  const int tx = threadIdx.x, ty = threadIdx.y;
  #pragma unroll
  for (int i = 0; i < 32; i += 8) {
    int hw = hw0 + tx;
    if (hw < HW_) tile[ty + i][tx] = src[(long long)(c0 + ty + i) * HW_ + hw];
  }
  __syncthreads();
  #pragma unroll
  for (int i = 0; i < 32; i += 8) {
    int hw = hw0 + ty + i;
    if (hw < HW_) dst[(long long)hw * C_ + (c0 + tx)] = tile[tx][ty + i];
  }
}

// Select element t (0..7) from a v8f with a cndmask chain (no scratch spill).
__device__ __forceinline__ float pm_sel8(v8f v, int t) {
  float d = v[0];
  d = (t == 1) ? v[1] : d;
  d = (t == 2) ? v[2] : d;
  d = (t == 3) ? v[3] : d;
  d = (t == 4) ? v[4] : d;
  d = (t == 5) ? v[5] : d;
  d = (t == 6) ? v[6] : d;
  d = (t == 7) ? v[7] : d;
  return d;
}

// Extract diag(D) of a 16x16 f32 C/D fragment: lanes 0..7 hold diag m=lane at
// vgpr[lane]; lanes 24..31 hold diag m=lane-16 at vgpr[lane-24]. Shuffle so
// lanes 0..15 end with diag(lane).
__device__ __forceinline__ float pm_diag(v8f acc, int lane) {
  int t = (lane < 8) ? lane : ((lane - 24) & 7);
  float d = pm_sel8(acc, t);
  int src = (lane < 8) ? lane : (lane + 16); // meaningful for lane<16
  return __shfl(d, src, 32);
}

// ---------------------------------------------------------------------------
// Phase 3: one wave per group of 16 prototypes. Gather each prototype's
// contiguous 512-f32 vector and accumulate three Gram diagonals with
// V_WMMA_F32_16X16X4_F32 (full f32 precision): pA.pB, pA.pA, pB.pB.
// A/B fragment layouts coincide for 16x16x4 f32 (lane -> index%16,
// k -> 2*(lane/16)+vgpr), so the same regs serve as A and B for the norms.
// grid = 2000 blocks * 4 waves = 8000 groups = 64 * (2000/16). EXEC always
// full (no divergence before the WMMAs), as the ISA requires.
// ---------------------------------------------------------------------------
__global__ void __launch_bounds__(128)
pm_proto_cos(const float* __restrict__ tA, const float* __restrict__ tB,
             const int* __restrict__ idxA, const int* __restrict__ idxB,
             float* __restrict__ cosOut) {
  const int wave = threadIdx.x >> 5;
  const int lane = threadIdx.x & 31;
  const int g = blockIdx.x * 4 + wave;        // 0..7999
  const int b = g / 125;
  const int p0 = (g % 125) * 16;
  const int m = lane & 15;                    // prototype within group
  const int khalf = lane >> 4;                // k sub-offset (0 or 1)*2

  const int ia = idxA[b * P_ + p0 + m];
  const int ib = idxB[b * P_ + p0 + m];
  const float* pa = tA + ((long long)b * HW_ + ia) * C_ + 2 * khalf;
  const float* pb = tB + ((long long)b * HW_ + ib) * C_ + 2 * khalf;

  v8f accAB = {};
  v8f accAA = {};
  v8f accBB = {};
  #pragma unroll 8
  for (int s = 0; s < 128; ++s) {             // K = 512 = 128 * 4
    v2f a = *(const v2f*)(pa + 4 * s);
    v2f c = *(const v2f*)(pb + 4 * s);
    accAB = __builtin_amdgcn_wmma_f32_16x16x4_f32(false, a, false, c,
                                                  (short)0, accAB, false, false);
    accAA = __builtin_amdgcn_wmma_f32_16x16x4_f32(false, a, false, a,
                                                  (short)0, accAA, false, false);
    accBB = __builtin_amdgcn_wmma_f32_16x16x4_f32(false, c, false, c,
                                                  (short)0, accBB, false, false);
  }

  float ab = pm_diag(accAB, lane);
  float aa = pm_diag(accAA, lane);
  float bb = pm_diag(accBB, lane);
  if (lane < 16) {
    float na = fmaxf(sqrtf(aa), EPS_);
    float nb = fmaxf(sqrtf(bb), EPS_);
    cosOut[b * P_ + p0 + lane] = ab / (na * nb);
  }
}

// ---------------------------------------------------------------------------
// Phase 4: logits[b] = sum_p cos[b,p]*w[p] + bias. One block per batch;
// fixed-order LDS reduction (deterministic, no float atomics).
// ---------------------------------------------------------------------------
__global__ void __launch_bounds__(256)
pm_logits(const float* __restrict__ cosv, const float* __restrict__ w,
          const float* __restrict__ bias, float* __restrict__ logits) {
  __shared__ float red[256];
  const int b = blockIdx.x;
  float s = 0.f;
  for (int p = threadIdx.x; p < P_; p += 256)
    s += cosv[b * P_ + p] * w[p];
  red[threadIdx.x] = s;
  __syncthreads();
  for (int o = 128; o > 0; o >>= 1) {
    if (threadIdx.x < o) red[threadIdx.x] += red[threadIdx.x + o];
    __syncthreads();
  }
  if (threadIdx.x == 0) logits[b] = red[0] + bias[0];
}

// ---------------------------------------------------------------------------
extern "C" void kernel_launch(void* const* d_in, const int* in_sizes, int n_in,
                              void* d_out, int out_size, void* d_ws, size_t ws_size,
                              hipStream_t stream) {
  const float* featA = (const float*)d_in[0];  // [B,C,H,W]
  const float* simA  = (const float*)d_in[1];  // [B,P,H,W]
  const float* featB = (const float*)d_in[2];
  const float* simB  = (const float*)d_in[3];
  const float* w     = (const float*)d_in[4];  // [1,P]
  const float* bias  = (const float*)d_in[5];  // [1]

  float* out = (float*)d_out;
  const long long BP = (long long)B_ * P_;     // 128000
  float* outLogits = out;                      // [64]
  float* outCos    = out + 64;                 // [B*P]
  float* outIA     = out + 64 + 1 * BP;
  float* outJA     = out + 64 + 2 * BP;
  float* outIB     = out + 64 + 3 * BP;
  float* outJB     = out + 64 + 4 * BP;

  // Workspace layout (16B-aligned sections)
  char* ws = (char*)d_ws;
  int* idxA = (int*)ws;                                    // 512000 B
  int* idxB = (int*)(ws + 512000);                         // 512000 B
  float* tA = (float*)(ws + 1024000);                      // 25,690,112 B
  float* tB = (float*)(ws + 1024000 + (size_t)B_ * HW_ * C_ * 4);

  // Phase 1: argmax over sim rows (2*128000 rows, 8 waves/block)
  {
    long long rows = 2 * BP;
    int grid = (int)((rows + 7) / 8);
    pm_argmax<<<grid, 256, 0, stream>>>(simA, simB, idxA, idxB,
                                        outIA, outJA, outIB, outJB);
  }
  // Phase 2: transpose feat to [B, HW, C]
  {
    dim3 grid((HW_ + 31) / 32, C_ / 32, 2 * B_);
    dim3 block(32, 8);
    pm_transpose<<<grid, block, 0, stream>>>(featA, featB, tA, tB);
  }
  // Phase 3: gather + WMMA Gram diagonals -> cos
  {
    pm_proto_cos<<<2000, 128, 0, stream>>>(tA, tB, idxA, idxB, outCos);
  }
  // Phase 4: logits
  {
    pm_logits<<<B_, 256, 0, stream>>>(outCos, w, bias, outLogits);
  }
}